// BezierReinforceWrapper_23321672417978
// MI455X (gfx1250) — compile-verified
//
#include <hip/hip_runtime.h>
#include <cmath>

// ---------------------------------------------------------------------------
// BezierReinforce: mu = x@W + b (WMMA f32), sigmoid -> sample, spline paint.
// Outputs (flat, in order): sketch[16384*784], log_prob[16384],
//                           entropy[16384], sample[16384*56]
// ---------------------------------------------------------------------------

typedef __attribute__((ext_vector_type(2))) float v2f;
typedef __attribute__((ext_vector_type(8))) float v8f;

#define B_ROWS   16384
#define K_DIM    512
#define N_DIM    56
#define NPADT    64          // N padded to 4 WMMA tiles
#define KC       128         // K chunk staged in LDS (64 K-pairs)
#define WP_STRIDE 80         // float2 stride per K-pair row (2*80 mod 64 == 32)
#define XSTRIDE  132         // float stride per M row (bank-conflict free b64)
#define ROWS_PB  64          // rows of x per block (4 waves x 16)
#define T1       128         // threads in GEMM kernel (4 wave32)

// log-prob / entropy constants: D * (-0.5*log(2pi)) and D * (0.5 + 0.5*log(2pi))
#define LOGP_CONST (-51.460557859461664f)
#define ENT_CONST  ( 79.460557859461664f)

__global__ __launch_bounds__(T1)
void gemm_sigmoid_kernel(const float* __restrict__ x,
                         const float* __restrict__ Wm,
                         const float* __restrict__ bias,
                         float* __restrict__ sample,
                         float* __restrict__ logp,
                         float* __restrict__ ent)
{
    __shared__ float ldsX[ROWS_PB * XSTRIDE];          // 64 rows x 128 K  (~34KB)
    __shared__ v2f   ldsWp[(KC / 2) * WP_STRIDE];      // 64 K-pairs x 64 N (40KB)

    const int tid     = threadIdx.x;
    const int lane    = tid & 31;
    const int wv      = tid >> 5;               // wave 0..3
    const int rowbase = blockIdx.x * ROWS_PB;

    // ISA f32 WMMA fragment geometry (wave32):
    //  A 16x4 : lanes 0-15 -> M=lane, K = k+0 / k+1 ; lanes 16-31 -> K = k+2 / k+3
    //  B 4x16 : row-striped across lanes, same K split as A
    //  C 16x16: VGPR r -> M=r (lanes 0-15) / M=r+8 (lanes 16-31), N = lane&15
    const int koff  = (lane < 16) ? 0 : 2;      // K offset within group-of-4
    const int khoff = (lane < 16) ? 0 : 1;      // K-pair offset within group-of-2
    const int m16   = lane & 15;
    const int nlo   = lane & 15;

    v8f acc[4] = {v8f{}, v8f{}, v8f{}, v8f{}};

    for (int kc = 0; kc < K_DIM; kc += KC) {
        // ---- stage W chunk as K-pairs: ldsWp[k/2][n] = {W[k][n], W[k+1][n]} ----
        for (int i = tid; i < (KC / 2) * NPADT; i += T1) {
            int kh = i >> 6, n = i & 63;
            v2f val = {0.0f, 0.0f};
            if (n < N_DIM) {
                val.x = Wm[(kc + 2 * kh + 0) * N_DIM + n];
                val.y = Wm[(kc + 2 * kh + 1) * N_DIM + n];
            }
            ldsWp[kh * WP_STRIDE + n] = val;
        }
        // ---- stage X chunk [ROWS_PB x KC], coalesced; prefetch next chunk ----
        for (int i = tid; i < ROWS_PB * KC; i += T1) {
            int r = i >> 7, c = i & (KC - 1);
            const float* src = &x[(size_t)(rowbase + r) * K_DIM + kc + c];
            ldsX[r * XSTRIDE + c] = *src;
            if (kc + KC < K_DIM) __builtin_prefetch(src + KC, 0, 1);
        }
        __syncthreads();

        const int arow = (wv * 16 + m16) * XSTRIDE;
        #pragma unroll 8
        for (int kk = 0; kk < KC; kk += 4) {
            // A fragment: contiguous K-pair -> single ds_load_b64
            v2f a = *reinterpret_cast<const v2f*>(&ldsX[arow + kk + koff]);
            const v2f* bp = &ldsWp[((kk >> 1) + khoff) * WP_STRIDE + nlo];
            #pragma unroll
            for (int nt = 0; nt < 4; ++nt) {
                v2f b = bp[nt * 16];            // contiguous pair -> ds_load_b64
                acc[nt] = __builtin_amdgcn_wmma_f32_16x16x4_f32(
                    /*neg_a=*/false, a, /*neg_b=*/false, b,
                    /*c_mod=*/(short)0, acc[nt],
                    /*reuse_a=*/false, /*reuse_b=*/false);
            }
        }
        __syncthreads();
    }

    // ---- epilogue: bias + sigmoid -> sample ----
    const int mbase = rowbase + wv * 16 + ((lane < 16) ? 0 : 8);
    #pragma unroll
    for (int nt = 0; nt < 4; ++nt) {
        int n = nt * 16 + nlo;
        if (n < N_DIM) {
            float bv = bias[n];
            #pragma unroll
            for (int r = 0; r < 8; ++r) {
                float mu = acc[nt][r] + bv;
                float sg = 1.0f / (1.0f + __expf(-mu));
                sample[(size_t)(mbase + r) * N_DIM + n] = sg;
            }
        }
    }

    // ---- per-row constants ----
    if (tid < ROWS_PB) {
        logp[rowbase + tid] = LOGP_CONST;
        ent[rowbase + tid]  = ENT_CONST;
    }
}

// ---------------------------------------------------------------------------
// Spline paint: one wave32 per image row, canvas in LDS, ds_add_f32 atomics.
// ---------------------------------------------------------------------------
#define ROWS_PB2 8
#define T2       256
#define CSTRIDE  792         // 784 padded

__global__ __launch_bounds__(T2)
void paint_kernel(const float* __restrict__ sample,
                  float* __restrict__ sketch)
{
    __shared__ float canvas[ROWS_PB2][CSTRIDE];
    __shared__ float sparams[ROWS_PB2][56];

    const int lane = threadIdx.x & 31;
    const int wv   = threadIdx.x >> 5;                 // 0..7
    const int row  = blockIdx.x * ROWS_PB2 + wv;

    for (int i = lane; i < 784; i += 32) canvas[wv][i] = 0.0f;
    for (int i = lane; i < 56; i += 32)
        sparams[wv][i] = sample[(size_t)row * N_DIM + i] * 28.0f;
    __syncthreads();

    // 8 splines x 50 t-samples = 400 points, strip-mined over 32 lanes
    for (int p = lane; p < 400; p += 32) {
        int s  = p / 50;
        int it = p - s * 50;
        float t  = (float)it * (1.0f / 49.0f);
        float u  = 1.0f - t;
        float c0 = u * u, c1 = 2.0f * u * t, c2 = t * t;
        const float* sp = &sparams[wv][s * 7];
        float px = c0 * sp[0] + c1 * sp[2] + c2 * sp[4];
        float py = c0 * sp[1] + c1 * sp[3] + c2 * sp[5];
        float wc = sp[6] * (-0.003f);
        int xi = (int)rintf(px);          // round-half-even, matches jnp.round
        int yi = (int)rintf(py);
        #pragma unroll
        for (int dx = -1; dx <= 1; ++dx) {
            int xc = xi + dx; xc = xc < 0 ? 0 : (xc > 27 ? 27 : xc);
            #pragma unroll
            for (int dy = -1; dy <= 1; ++dy) {
                int yc = yi + dy; yc = yc < 0 ? 0 : (yc > 27 ? 27 : yc);
                atomicAdd(&canvas[wv][xc * 28 + yc], wc);   // ds_add_f32
            }
        }
    }
    __syncthreads();

    // +0.3, clamp [0,1], coalesced store
    for (int i = lane; i < 784; i += 32) {
        float v = canvas[wv][i] + 0.3f;
        v = fminf(fmaxf(v, 0.0f), 1.0f);
        sketch[(size_t)row * 784 + i] = v;
    }
}

// ---------------------------------------------------------------------------
extern "C" void kernel_launch(void* const* d_in, const int* in_sizes, int n_in,
                              void* d_out, int out_size, void* d_ws, size_t ws_size,
                              hipStream_t stream) {
    const float* x  = (const float*)d_in[0];   // [16384, 512]
    const float* Wm = (const float*)d_in[1];   // [512, 56]
    const float* b  = (const float*)d_in[2];   // [56]

    float* out = (float*)d_out;
    const size_t NB = B_ROWS;
    float* sketch = out;                        // [16384, 784]
    float* logp   = out + NB * 784;             // [16384]
    float* ent    = logp + NB;                  // [16384]
    float* sample = ent + NB;                   // [16384, 56]

    dim3 g1(B_ROWS / ROWS_PB);                  // 256 blocks
    gemm_sigmoid_kernel<<<g1, T1, 0, stream>>>(x, Wm, b, sample, logp, ent);

    dim3 g2(B_ROWS / ROWS_PB2);                 // 2048 blocks
    paint_kernel<<<g2, T2, 0, stream>>>(sample, sketch);
}